// EnhancedSynGCN_66632122630754
// MI455X (gfx1250) — compile-verified
//
#include <hip/hip_runtime.h>
#include <hip/hip_bf16.h>

// Problem constants (from reference): B,N,D,H,L,K = 8,128,768,4,2,3
constexpr int cB = 8, cN = 128, cD = 768, cH = 4, cHD = 192, cL = 2, cK = 3;
constexpr int cBN   = cB * cN;            // 1024
constexpr int cBND  = cB * cN * cD;       // 786432
constexpr int cBNN  = cB * cN * cN;       // 131072
constexpr int cBNNH = cB * cN * cN * cH;  // 524288

typedef float v2f __attribute__((ext_vector_type(2)));
typedef float v8f __attribute__((ext_vector_type(8)));

// ---------------------------------------------------------------------------
// Generic strided/batched WMMA fp32 GEMM:
//   C[z,m,n] = act( alpha * sum_k A[z,m,k] * (W[z,k,n] - W2[z,k,n]?) (* rowScale[z,m]?)
//              + bias[n]? + beta * Cin[z,m,n]? )
// z = blockIdx.z split as z1 = z / ZH, z2 = z % ZH (two-level batch for (b,h)).
// Block = 128 threads = 4 waves along M; each wave owns a 32x64 tile:
// 2 M-subtiles x 4 N-subtiles of 16x16 V_WMMA_F32_16X16X4_F32 accumulators.
// A fragment reused across 4 N-tiles, loaded as one b64 when K-stride==1 (A64).
// ---------------------------------------------------------------------------
struct GP {
  const float* A; int lda; int sA; long sAz1; long sAz2;
  const float* W; int ldw; long sWz1; long sWz2;
  const float* W2;
  const float* bias;
  const float* Cin; float beta;
  const float* rowScale; int rsStride;
  float alpha;
  float* C; int ldc; long sCz1; long sCz2;
  int M, K, N, ZH;
};

template <int ACT, bool WSUB, bool A64>
__global__ __launch_bounds__(128) void wmma_gemm_k(GP p) {
  const int lane = threadIdx.x & 31;
  const int wid  = threadIdx.x >> 5;
  const int tm = blockIdx.y * 128 + wid * 32;
  const int tn = blockIdx.x * 64;
  const int z  = blockIdx.z;
  const int z1 = z / p.ZH;
  const int z2 = z - z1 * p.ZH;

  const int rm = lane & 15;        // M-row (A) / N-col (B,C) within tile
  const int kh = (lane >> 4) << 1; // lanes 16-31 own K+2,K+3

  const long aoff = (long)z1 * p.sAz1 + (long)z2 * p.sAz2 + (long)kh * p.sA;
  const float* Ap0 = p.A + aoff + (long)(tm + rm) * p.lda;
  const float* Ap1 = p.A + aoff + (long)(tm + 16 + rm) * p.lda;
  const long woff = (long)z1 * p.sWz1 + (long)z2 * p.sWz2 +
                    (long)kh * p.ldw + (tn + rm);
  const float* Wp  = p.W + woff;
  const float* W2p = WSUB ? (p.W2 + woff) : nullptr;

  v8f acc[2][4];
#pragma unroll
  for (int mt = 0; mt < 2; ++mt)
#pragma unroll
    for (int nt = 0; nt < 4; ++nt)
      acc[mt][nt] = v8f{0.f, 0.f, 0.f, 0.f, 0.f, 0.f, 0.f, 0.f};

#pragma unroll 2
  for (int k = 0; k < p.K; k += 4) {
    v2f a0, a1;
    if (A64) {                       // contiguous K: single 8-byte load per frag
      a0 = *(const v2f*)Ap0;
      a1 = *(const v2f*)Ap1;
    } else {                         // strided K (attention alpha, stride H)
      a0.x = Ap0[0]; a0.y = Ap0[p.sA];
      a1.x = Ap1[0]; a1.y = Ap1[p.sA];
    }
    v2f b[4];
#pragma unroll
    for (int nt = 0; nt < 4; ++nt) {
      b[nt].x = Wp[nt * 16];
      b[nt].y = Wp[p.ldw + nt * 16];
      if (WSUB) {
        b[nt].x -= W2p[nt * 16];
        b[nt].y -= W2p[p.ldw + nt * 16];
      }
    }
#pragma unroll
    for (int nt = 0; nt < 4; ++nt) {
      acc[0][nt] = __builtin_amdgcn_wmma_f32_16x16x4_f32(
          false, a0, false, b[nt], (short)0, acc[0][nt], false, false);
      acc[1][nt] = __builtin_amdgcn_wmma_f32_16x16x4_f32(
          false, a1, false, b[nt], (short)0, acc[1][nt], false, false);
    }
    Ap0 += 4L * p.sA;
    Ap1 += 4L * p.sA;
    Wp  += 4L * p.ldw;
    if (WSUB) W2p += 4L * p.ldw;
  }

  const int h16 = lane >> 4;
#pragma unroll
  for (int mt = 0; mt < 2; ++mt) {
#pragma unroll
    for (int nt = 0; nt < 4; ++nt) {
      const int col = tn + nt * 16 + rm;
      const float bi = p.bias ? p.bias[col] : 0.f;
      const long cb = (long)z1 * p.sCz1 + (long)z2 * p.sCz2 + col;
#pragma unroll
      for (int r = 0; r < 8; ++r) {
        const int row = tm + mt * 16 + r + 8 * h16;
        float v = acc[mt][nt][r] * p.alpha;
        if (p.rowScale) v *= p.rowScale[z1 * p.rsStride + row];
        v += bi;
        if (p.Cin) v += p.beta * p.Cin[cb + (long)row * p.ldc];
        if (ACT == 1) v = fmaxf(v, 0.f);
        else if (ACT == 2) v = v > 0.f ? v : (__expf(v) - 1.f);
        p.C[cb + (long)row * p.ldc] = v;
      }
    }
  }
}

// ---------------------------------------------------------------------------
// Degree sums: rows[b,i] = sum_j adj, rinv = 1/(rows+1e-6), cols[b,j] = sum_i adj
// ---------------------------------------------------------------------------
__global__ void rowcol_k(const float* __restrict__ adj, float* rows,
                         float* rinv, float* cols) {
  int idx = blockIdx.x * blockDim.x + threadIdx.x;
  if (idx < cBN) {
    const float* r = adj + (long)idx * cN;
    float s = 0.f;
    for (int j = 0; j < cN; ++j) s += r[j];
    rows[idx] = s;
    rinv[idx] = 1.f / (s + 1e-6f);
  } else if (idx < 2 * cBN) {
    int t = idx - cBN;
    int b = t >> 7, j = t & (cN - 1);
    float s = 0.f;
    for (int i = 0; i < cN; ++i) s += adj[((long)b * cN + i) * cN + j];
    cols[t] = s;
  }
}

// L_norm[b,i,j] = I - d_i * adj * d_j,  d = rsqrt(rowsum + 1e-6)
__global__ void lnorm_k(const float* __restrict__ adj,
                        const float* __restrict__ rows, float* Ln) {
  int idx = blockIdx.x * blockDim.x + threadIdx.x;
  if (idx >= cBNN) return;
  int j  = idx & (cN - 1);
  int bi = idx >> 7;
  int b  = bi >> 7;
  int i  = bi & (cN - 1);
  float di = rsqrtf(rows[bi] + 1e-6f);
  float dj = rsqrtf(rows[b * cN + j] + 1e-6f);
  float v = -di * adj[idx] * dj;
  if (i == j) v += 1.f;
  Ln[idx] = v;
}

// GAT s/t projections: s[b,n,h] = <h[b,n,h,:], a[h,:HD]>, t uses a[h,HD:]
__global__ void gat_st_k(const float* __restrict__ h,
                         const float* __restrict__ ga, float* s, float* t) {
  int idx = blockIdx.x * blockDim.x + threadIdx.x;
  if (idx >= cBN * cH) return;
  int hh = idx & (cH - 1);
  int bn = idx >> 2;
  const float* hp = h + (long)bn * cD + hh * cHD;
  const float* av = ga + hh * 2 * cHD;
  float ss = 0.f, tt = 0.f;
  for (int d = 0; d < cHD; ++d) {
    float v = hp[d];
    ss += v * av[d];
    tt += v * av[cHD + d];
  }
  s[idx] = ss;
  t[idx] = tt;
}

// One wave32 per (b,i,h): leaky + mask + softmax over 128 neighbors.
__global__ void gat_softmax_k(const float* __restrict__ s,
                              const float* __restrict__ t,
                              const float* __restrict__ adj,
                              float* __restrict__ alpha) {
  int warp = (blockIdx.x * blockDim.x + threadIdx.x) >> 5;
  int lane = threadIdx.x & 31;
  if (warp >= cBN * cH) return;
  int hh = warp & (cH - 1);
  int bi = warp >> 2;          // b*N + i
  int b  = bi >> 7;
  float si = s[bi * cH + hh];
  const float* trow = t + (long)b * cN * cH + hh;
  const float* arow = adj + (long)bi * cN;
  float e[4];
  float m = -1e30f;
#pragma unroll
  for (int q = 0; q < 4; ++q) {
    int j = lane + q * 32;
    float ev = si + trow[j * cH];
    ev = ev < 0.f ? 0.2f * ev : ev;
    if (arow[j] == 0.f) ev = -1e9f;
    e[q] = ev;
    m = fmaxf(m, ev);
  }
  for (int o = 16; o > 0; o >>= 1) m = fmaxf(m, __shfl_xor(m, o, 32));
  float sum = 0.f;
#pragma unroll
  for (int q = 0; q < 4; ++q) { e[q] = __expf(e[q] - m); sum += e[q]; }
  for (int o = 16; o > 0; o >>= 1) sum += __shfl_xor(sum, o, 32);
  float inv = 1.f / sum;
#pragma unroll
  for (int q = 0; q < 4; ++q) {
    int j = lane + q * 32;
    alpha[((long)bi * cN + j) * cH + hh] = e[q] * inv;
  }
}

// Edge branch: out[b,i,d] = max_j ( adj ? relu(pre_i[b,i,d]+pre_j[b,j,d]) : 0 )
// (never materializes the (B,N,N,D) pair tensor)
__global__ void edge_max_k(const float* __restrict__ pre_i,
                           const float* __restrict__ pre_j,
                           const float* __restrict__ adj,
                           float* __restrict__ out) {
  int d  = blockIdx.x * blockDim.x + threadIdx.x;  // 0..767
  int bn = blockIdx.y;                              // b*N+i
  int b  = bn >> 7;
  __shared__ float arow[cN];
  for (int j = threadIdx.x; j < cN; j += blockDim.x)
    arow[j] = adj[(long)bn * cN + j];
  __syncthreads();
  float pi = pre_i[(long)bn * cD + d];
  const float* pj = pre_j + (long)b * cN * cD + d;
  float m = 0.f;
  for (int j = 0; j < cN; ++j) {
    if (arow[j] != 0.f) {
      float v = fmaxf(pi + pj[(long)j * cD], 0.f);
      m = fmaxf(m, v);
    }
  }
  out[(long)bn * cD + d] = m;
}

// Fusion attention: per (b,n) wave: branch means -> 4x4 matmul -> softmax -> w
__global__ void fuse_scores_k(const float* g0, const float* g1, const float* g2,
                              const float* g3, const float* __restrict__ fW,
                              const float* __restrict__ fb, float* wout) {
  int warp = (blockIdx.x * blockDim.x + threadIdx.x) >> 5;
  int lane = threadIdx.x & 31;
  if (warp >= cBN) return;
  const float* bufs[4] = {g0, g1, g2, g3};
  float mean[4];
#pragma unroll
  for (int c = 0; c < 4; ++c) {
    const float* p = bufs[c] + (long)warp * cD;
    float sum = 0.f;
    for (int d = lane; d < cD; d += 32) sum += p[d];
    for (int o = 16; o > 0; o >>= 1) sum += __shfl_xor(sum, o, 32);
    mean[c] = sum * (1.f / (float)cD);
  }
  float sc[4], mx = -1e30f;
#pragma unroll
  for (int c2 = 0; c2 < 4; ++c2) {
    float v = fb[c2];
#pragma unroll
    for (int c = 0; c < 4; ++c) v += mean[c] * fW[c * 4 + c2];
    sc[c2] = v;
    mx = fmaxf(mx, v);
  }
  float sum = 0.f;
#pragma unroll
  for (int c2 = 0; c2 < 4; ++c2) { sc[c2] = __expf(sc[c2] - mx); sum += sc[c2]; }
  float inv = 1.f / sum;
  if (lane < 4) wout[warp * 4 + lane] = sc[lane] * inv;
}

__global__ void fuse_apply_k(const float* g0, const float* g1, const float* g2,
                             const float* g3, const float* __restrict__ w,
                             float* fused) {
  int idx = blockIdx.x * blockDim.x + threadIdx.x;
  if (idx >= cBND) return;
  int bn = idx / cD;
  const float* wp = w + bn * 4;
  fused[idx] = wp[0] * g0[idx] + wp[1] * g1[idx] + wp[2] * g2[idx] + wp[3] * g3[idx];
}

__global__ void mask_k(const float* rows, const float* cols, float* out) {
  int idx = blockIdx.x * blockDim.x + threadIdx.x;
  if (idx >= cBN) return;
  out[idx] = (rows[idx] + cols[idx] == 0.f) ? 1.f : 0.f;
}

// ---------------------------------------------------------------------------
// Host-side GEMM dispatcher
// ---------------------------------------------------------------------------
static void run_gemm(hipStream_t st, const float* A, int lda, int sA, long sAz1,
                     long sAz2, const float* W, int ldw, long sWz1, long sWz2,
                     const float* W2, const float* bias, const float* Cin,
                     float beta, const float* rowScale, int rsStride,
                     float alpha, float* C, int ldc, long sCz1, long sCz2,
                     int M, int K, int N, int Z, int ZH, int act) {
  GP p{A, lda, sA, sAz1, sAz2, W, ldw, sWz1, sWz2, W2, bias, Cin, beta,
       rowScale, rsStride, alpha, C, ldc, sCz1, sCz2, M, K, N, ZH};
  dim3 g(N / 64, M / 128, Z), b(128);
  const bool a64 = (sA == 1);
  if (W2) {
    if (a64) wmma_gemm_k<0, true, true><<<g, b, 0, st>>>(p);
    else     wmma_gemm_k<0, true, false><<<g, b, 0, st>>>(p);
  } else if (act == 0) {
    if (a64) wmma_gemm_k<0, false, true><<<g, b, 0, st>>>(p);
    else     wmma_gemm_k<0, false, false><<<g, b, 0, st>>>(p);
  } else if (act == 1) {
    if (a64) wmma_gemm_k<1, false, true><<<g, b, 0, st>>>(p);
    else     wmma_gemm_k<1, false, false><<<g, b, 0, st>>>(p);
  } else {
    if (a64) wmma_gemm_k<2, false, true><<<g, b, 0, st>>>(p);
    else     wmma_gemm_k<2, false, false><<<g, b, 0, st>>>(p);
  }
}

// Plain GEMM: C[M,N] = act(A[M,K] @ W + bias + beta*Cin)
static void gemm2d(hipStream_t st, const float* A, const float* W,
                   const float* W2, const float* bias, const float* Cin,
                   float beta, float* C, int M, int K, int N, int act) {
  run_gemm(st, A, K, 1, 0, 0, W, N, 0, 0, W2, bias, Cin, beta, nullptr, 0, 1.f,
           C, N, 0, 0, M, K, N, 1, 1, act);
}

extern "C" void kernel_launch(void* const* d_in, const int* in_sizes, int n_in,
                              void* d_out, int out_size, void* d_ws,
                              size_t ws_size, hipStream_t stream) {
  (void)in_sizes; (void)n_in; (void)out_size; (void)ws_size;
  int p = 0;
  const float* adj = (const float*)d_in[p++];
  const float* xin = (const float*)d_in[p++];
  const float* gat_W[cL]; for (int l = 0; l < cL; ++l) gat_W[l] = (const float*)d_in[p++];
  const float* gat_a[cL]; for (int l = 0; l < cL; ++l) gat_a[l] = (const float*)d_in[p++];
  const float* sage_W[cL]; for (int l = 0; l < cL; ++l) sage_W[l] = (const float*)d_in[p++];
  const float* sage_b[cL]; for (int l = 0; l < cL; ++l) sage_b[l] = (const float*)d_in[p++];
  const float* cheb_W[cL][cK];
  for (int l = 0; l < cL; ++l) for (int k = 0; k < cK; ++k) cheb_W[l][k] = (const float*)d_in[p++];
  const float* cheb_b[cL][cK];
  for (int l = 0; l < cL; ++l) for (int k = 0; k < cK; ++k) cheb_b[l][k] = (const float*)d_in[p++];
  const float* edge_W[cL]; for (int l = 0; l < cL; ++l) edge_W[l] = (const float*)d_in[p++];
  const float* edge_b[cL]; for (int l = 0; l < cL; ++l) edge_b[l] = (const float*)d_in[p++];
  const float* fattn_W = (const float*)d_in[p++];
  const float* fattn_b = (const float*)d_in[p++];
  const float* ffuse_W = (const float*)d_in[p++];
  const float* ffuse_b = (const float*)d_in[p++];

  float* wsf   = (float*)d_ws;
  float* gatF  = wsf;
  float* sageF = gatF + cBND;
  float* chebF = sageF + cBND;
  float* edgeF = chebF + cBND;
  float* bufA  = edgeF + cBND;
  float* bufB  = bufA + cBND;
  float* bufC  = bufB + cBND;
  float* attn  = bufC + cBND;         // B*N*N*H
  float* lnorm = attn + cBNNH;        // B*N*N
  float* rows  = lnorm + cBNN;
  float* rinv  = rows + cBN;
  float* cols  = rinv + cBN;
  float* sbuf  = cols + cBN;          // B*N*H
  float* tbuf  = sbuf + cBN * cH;
  float* wfuse = tbuf + cBN * cH;     // B*N*4

  rowcol_k<<<8, 256, 0, stream>>>(adj, rows, rinv, cols);
  lnorm_k<<<cBNN / 256, 256, 0, stream>>>(adj, rows, lnorm);

  // ---------------- GAT branch ----------------
  {
    const float* gx = xin;
    for (int l = 0; l < cL; ++l) {
      float* dest = (l == cL - 1) ? gatF : bufA;
      gemm2d(stream, gx, gat_W[l], nullptr, nullptr, nullptr, 0.f, bufB,
             cBN, cD, cD, 0);                               // h = x @ W
      gat_st_k<<<16, 256, 0, stream>>>(bufB, gat_a[l], sbuf, tbuf);
      gat_softmax_k<<<512, 256, 0, stream>>>(sbuf, tbuf, adj, attn);
      // hp[b,:,h,:] = elu( alpha[b,:,:,h] @ hmat[b,:,h,:] ), batched over (b,h)
      run_gemm(stream, attn, cN * cH, cH, (long)cN * cN * cH, 1,
               bufB, cD, (long)cN * cD, cHD, nullptr, nullptr, nullptr, 0.f,
               nullptr, 0, 1.f, dest, cD, (long)cN * cD, cHD,
               cN, cN, cHD, cB * cH, cH, /*elu*/ 2);
      gx = dest;
    }
  }
  // ---------------- SAGE branch ----------------
  {
    const float* sx = xin;
    for (int l = 0; l < cL; ++l) {
      float* dest = (l == cL - 1) ? sageF : bufA;
      // neigh = (adj @ x) * rinv_row
      run_gemm(stream, adj, cN, 1, (long)cN * cN, 0, sx, cD, (long)cN * cD, 0,
               nullptr, nullptr, nullptr, 0.f, rinv, cN, 1.f, bufB, cD,
               (long)cN * cD, 0, cN, cN, cD, cB, 1, 0);
      // out = relu(x @ Wtop + neigh @ Wbot + b)   (concat GEMM split in two)
      gemm2d(stream, sx, sage_W[l], nullptr, nullptr, nullptr, 0.f, dest,
             cBN, cD, cD, 0);
      gemm2d(stream, bufB, sage_W[l] + (long)cD * cD, nullptr, sage_b[l], dest,
             1.f, dest, cBN, cD, cD, 1);
      sx = dest;
    }
  }
  // ---------------- Chebyshev branch ----------------
  {
    const float* cx = xin;
    for (int l = 0; l < cL; ++l) {
      float* dest = (l == cL - 1) ? chebF : bufA;
      // Tx1 = L @ x
      run_gemm(stream, lnorm, cN, 1, (long)cN * cN, 0, cx, cD, (long)cN * cD, 0,
               nullptr, nullptr, nullptr, 0.f, nullptr, 0, 1.f, bufB, cD,
               (long)cN * cD, 0, cN, cN, cD, cB, 1, 0);
      gemm2d(stream, cx, cheb_W[l][0], nullptr, cheb_b[l][0], nullptr, 0.f,
             dest, cBN, cD, cD, 0);                          // out  = Tx0@W0+b0
      gemm2d(stream, bufB, cheb_W[l][1], nullptr, cheb_b[l][1], dest, 1.f,
             dest, cBN, cD, cD, 0);                          // out += Tx1@W1+b1
      // Tx2 = 2*L@Tx1 - Tx0
      run_gemm(stream, lnorm, cN, 1, (long)cN * cN, 0, bufB, cD, (long)cN * cD,
               0, nullptr, nullptr, cx, -1.f, nullptr, 0, 2.f, bufC, cD,
               (long)cN * cD, 0, cN, cN, cD, cB, 1, 0);
      gemm2d(stream, bufC, cheb_W[l][2], nullptr, cheb_b[l][2], dest, 1.f,
             dest, cBN, cD, cD, 1);                          // out += Tx2@W2+b2, relu
      cx = dest;
    }
  }
  // ---------------- Edge branch ----------------
  {
    const float* ex = xin;
    for (int l = 0; l < cL; ++l) {
      float* dest = (l == cL - 1) ? edgeF : bufA;
      // pre_i = x @ (Wa - Wb) + b ; pre_j = x @ Wb
      gemm2d(stream, ex, edge_W[l], edge_W[l] + (long)cD * cD, edge_b[l],
             nullptr, 0.f, bufB, cBN, cD, cD, 0);
      gemm2d(stream, ex, edge_W[l] + (long)cD * cD, nullptr, nullptr, nullptr,
             0.f, bufC, cBN, cD, cD, 0);
      edge_max_k<<<dim3(cD / 256, cBN), 256, 0, stream>>>(bufB, bufC, adj, dest);
      ex = dest;
    }
  }
  // ---------------- Fusion ----------------
  fuse_scores_k<<<128, 256, 0, stream>>>(gatF, sageF, chebF, edgeF, fattn_W,
                                         fattn_b, wfuse);
  fuse_apply_k<<<cBND / 256, 256, 0, stream>>>(gatF, sageF, chebF, edgeF,
                                               wfuse, bufA);
  gemm2d(stream, bufA, ffuse_W, nullptr, ffuse_b, nullptr, 0.f, (float*)d_out,
         cBN, cD, cD, 0);
  mask_k<<<4, 256, 0, stream>>>(rows, cols, (float*)d_out + cBND);
}